// RNN_59382217834631
// MI455X (gfx1250) — compile-verified
//
#include <hip/hip_runtime.h>
#include <hip/hip_bf16.h>

// Problem constants (match reference)
constexpr int T  = 512;
constexpr int B  = 256;
constexpr int N  = 1024;
constexpr int NI = 128;
constexpr int NO = 64;
#define ALPHA 0.2f

typedef float v2f __attribute__((ext_vector_type(2)));
typedef float v8f __attribute__((ext_vector_type(8)));

__device__ __forceinline__ v2f ld2(const float* p) { return *(const v2f*)p; }

// ---------------------------------------------------------------------------
// Weight packing: P[(k>>1)*2*ldn + 2*n + (k&1)] = W[k][n]
// makes each WMMA B-fragment (pair of K rows for one column) a contiguous
// 8B-aligned float2 load.
// ---------------------------------------------------------------------------

// Whh_eff[k][n] = mask[k][n] ? W_hh[n][k] : 0   (mask * W_hh^T), packed
__global__ void pack_whh_kernel(const float* __restrict__ whh,
                                const unsigned char* __restrict__ mask,
                                float* __restrict__ dst) {
  int idx = blockIdx.x * blockDim.x + threadIdx.x;
  if (idx >= N * N) return;
  int k = idx / N, n = idx % N;
  float v = mask[(size_t)k * N + n] ? whh[(size_t)n * N + k] : 0.0f;
  dst[(size_t)(k >> 1) * (2 * N) + 2 * n + (k & 1)] = v;
}

// Wih_eff[k][n] = W_ih[n][k]  (W_ih^T), packed. k in [0,NI), n in [0,N)
__global__ void pack_wih_kernel(const float* __restrict__ wih,
                                float* __restrict__ dst) {
  int idx = blockIdx.x * blockDim.x + threadIdx.x;
  if (idx >= NI * N) return;
  int k = idx / N, n = idx % N;
  dst[(size_t)(k >> 1) * (2 * N) + 2 * n + (k & 1)] = wih[(size_t)n * NI + k];
}

// Wout_T[n][o] = W_out[o][n], packed with ldn = NO. n in [0,N), o in [0,NO)
__global__ void pack_wout_kernel(const float* __restrict__ wout,
                                 float* __restrict__ dst) {
  int idx = blockIdx.x * blockDim.x + threadIdx.x;
  if (idx >= N * NO) return;
  int n = idx / NO, o = idx % NO;
  dst[(size_t)(n >> 1) * (2 * NO) + 2 * o + (n & 1)] = wout[(size_t)o * N + n];
}

// x = h0 ; r0 = tanh(h0)
__global__ void init_state_kernel(const float* __restrict__ h0,
                                  float* __restrict__ x,
                                  float* __restrict__ r0) {
  int idx = blockIdx.x * blockDim.x + threadIdx.x;
  if (idx >= B * N) return;
  float v = h0[idx];
  x[idx]  = v;
  r0[idx] = tanhf(v);
}

// ---------------------------------------------------------------------------
// Core f32 WMMA tile: one wave accumulates a 32x32 tile as 2x2 C tiles of
// V_WMMA_F32_16X16X4_F32. A is row-major [M][lda]; Bp is K-pair-packed.
// ---------------------------------------------------------------------------
__device__ __forceinline__ void gemm_tile_f32(const float* __restrict__ A, int lda,
                                              const float* __restrict__ Bp, int ldn,
                                              int K, int mBase, int nBase, int lane,
                                              v8f (&c)[2][2]) {
  const int l    = lane & 15;
  const int half = lane >> 4;
  const float* a0p = A + (size_t)(mBase + l) * lda + 2 * half;
  const float* a1p = A + (size_t)(mBase + 16 + l) * lda + 2 * half;
  const float* b0p = Bp + 2 * (nBase + l);
  const float* b1p = Bp + 2 * (nBase + 16 + l);
  const int ld2n = 2 * ldn;
#pragma unroll 4
  for (int k0 = 0; k0 < K; k0 += 4) {
    v2f a0 = ld2(a0p + k0);
    v2f a1 = ld2(a1p + k0);
    int boff = ((k0 >> 1) + half) * ld2n;
    v2f b0 = ld2(b0p + boff);
    v2f b1 = ld2(b1p + boff);
    c[0][0] = __builtin_amdgcn_wmma_f32_16x16x4_f32(false, a0, false, b0, (short)0, c[0][0], false, false);
    c[0][1] = __builtin_amdgcn_wmma_f32_16x16x4_f32(false, a0, false, b1, (short)0, c[0][1], false, false);
    c[1][0] = __builtin_amdgcn_wmma_f32_16x16x4_f32(false, a1, false, b0, (short)0, c[1][0], false, false);
    c[1][1] = __builtin_amdgcn_wmma_f32_16x16x4_f32(false, a1, false, b1, (short)0, c[1][1], false, false);
  }
}

// ---------------------------------------------------------------------------
// One RNN step: acc = rprev @ Whh_m + x_t @ Wih_T ; x += alpha*(acc - x)
// r_t = tanh(x) -> written straight into hiddenl1[t] slab (rout).
// Grid: (N/64, B/64), block 128 (4 waves, 64x64 WG tile, 32x32 per wave).
// ---------------------------------------------------------------------------
__global__ void __launch_bounds__(128) rnn_step_kernel(
    const float* __restrict__ rprev,  // [B][N]
    float* __restrict__ x,            // [B][N] (in-place)
    const float* __restrict__ xt,     // [B][NI]
    const float* __restrict__ whh_p,  // packed [N][N]
    const float* __restrict__ wih_p,  // packed [NI][N]
    float* __restrict__ rout)         // [B][N]
{
  const int lane = threadIdx.x & 31;
  const int wave = threadIdx.x >> 5;
  const int mBase = blockIdx.y * 64 + (wave >> 1) * 32;  // batch rows
  const int nBase = blockIdx.x * 64 + (wave & 1) * 32;   // hidden cols

  v8f c[2][2] = {};
  gemm_tile_f32(rprev, N, whh_p, N, N,  mBase, nBase, lane, c);
  gemm_tile_f32(xt,   NI, wih_p, N, NI, mBase, nBase, lane, c);

  const int l    = lane & 15;
  const int half = lane >> 4;
#pragma unroll
  for (int mt = 0; mt < 2; ++mt) {
#pragma unroll
    for (int nt = 0; nt < 2; ++nt) {
      const int col = nBase + nt * 16 + l;
#pragma unroll
      for (int v = 0; v < 8; ++v) {
        const int row = mBase + mt * 16 + half * 8 + v;
        const size_t idx = (size_t)row * N + col;
        float xo = x[idx];
        float xn = fmaf(ALPHA, c[mt][nt][v] - xo, xo);  // x + a*(-x + acc)
        x[idx]    = xn;
        rout[idx] = tanhf(xn);
      }
    }
  }
}

// ---------------------------------------------------------------------------
// Per-step output projection: out = r_t @ Wout_T + b_out.  Grid: (NO/64, B/64)
// ---------------------------------------------------------------------------
__global__ void __launch_bounds__(128) rnn_out_kernel(
    const float* __restrict__ r,       // [B][N]
    const float* __restrict__ wout_p,  // packed [N][NO]
    const float* __restrict__ bias,    // [NO]
    float* __restrict__ out)           // [B][NO]
{
  const int lane = threadIdx.x & 31;
  const int wave = threadIdx.x >> 5;
  const int mBase = blockIdx.y * 64 + (wave >> 1) * 32;
  const int nBase = blockIdx.x * 64 + (wave & 1) * 32;

  v8f c[2][2] = {};
  gemm_tile_f32(r, N, wout_p, NO, N, mBase, nBase, lane, c);

  const int l    = lane & 15;
  const int half = lane >> 4;
#pragma unroll
  for (int mt = 0; mt < 2; ++mt) {
#pragma unroll
    for (int nt = 0; nt < 2; ++nt) {
      const int col = nBase + nt * 16 + l;
      const float bv = bias[col];
#pragma unroll
      for (int v = 0; v < 8; ++v) {
        const int row = mBase + mt * 16 + half * 8 + v;
        out[(size_t)row * NO + col] = c[mt][nt][v] + bv;
      }
    }
  }
}

// ---------------------------------------------------------------------------
extern "C" void kernel_launch(void* const* d_in, const int* in_sizes, int n_in,
                              void* d_out, int out_size, void* d_ws, size_t ws_size,
                              hipStream_t stream) {
  const float* X     = (const float*)d_in[0];          // [T][B][NI]
  const float* h0    = (const float*)d_in[1];          // [B][N]
  const float* W_ih  = (const float*)d_in[2];          // [N][NI]
  const float* W_hh  = (const float*)d_in[3];          // [N][N]
  const float* W_out = (const float*)d_in[4];          // [NO][N]
  const float* b_out = (const float*)d_in[5];          // [NO]
  const unsigned char* mask = (const unsigned char*)d_in[6];  // jax bool: 1B/elem

  // Workspace layout (floats): packed weights + state (~6.75 MB)
  float* ws     = (float*)d_ws;
  float* whh_p  = ws;                            // N*N
  float* wih_p  = whh_p + (size_t)N * N;         // NI*N
  float* wout_p = wih_p + (size_t)NI * N;        // N*NO
  float* xstate = wout_p + (size_t)N * NO;       // B*N
  float* r0     = xstate + (size_t)B * N;        // B*N

  // d_out: [T*B*NO] out, then [T*B*N] hiddenl1
  float* out = (float*)d_out;
  float* hid = out + (size_t)T * B * NO;

  pack_whh_kernel<<<(N * N + 255) / 256, 256, 0, stream>>>(W_hh, mask, whh_p);
  pack_wih_kernel<<<(NI * N + 255) / 256, 256, 0, stream>>>(W_ih, wih_p);
  pack_wout_kernel<<<(N * NO + 255) / 256, 256, 0, stream>>>(W_out, wout_p);
  init_state_kernel<<<(B * N + 255) / 256, 256, 0, stream>>>(h0, xstate, r0);

  for (int t = 0; t < T; ++t) {
    const float* rprev = (t == 0) ? r0 : (hid + (size_t)(t - 1) * B * N);
    rnn_step_kernel<<<dim3(N / 64, B / 64), 128, 0, stream>>>(
        rprev, xstate, X + (size_t)t * B * NI, whh_p, wih_p,
        hid + (size_t)t * B * N);
    rnn_out_kernel<<<dim3(NO / 64, B / 64), 128, 0, stream>>>(
        hid + (size_t)t * B * N, wout_p, b_out, out + (size_t)t * B * NO);
  }
}